// UtoU_6597069767211
// MI455X (gfx1250) — compile-verified
//
#include <hip/hip_runtime.h>
#include <stdint.h>

typedef float v2f __attribute__((ext_vector_type(2)));
typedef float v8f __attribute__((ext_vector_type(8)));

namespace {
constexpr int kC = 32;             // colors
constexpr int kD = 128;            // channels
constexpr int kNPG = 4;            // nodes per group (128 flat rows)
constexpr int kRows = kNPG * kC;   // 128
constexpr int kStride = 132;       // padded LDS row stride (bank-conflict-free, 16B aligned)
}

// 16-byte Global -> LDS async copy (ASYNCcnt-tracked), via CDNA5 ISA mnemonic.
// VDST = LDS byte offset (low 32 bits of the flat shared pointer), VADDR = 64-bit
// global address pair. Both must be 16B aligned.
__device__ __forceinline__ void copy16_to_lds(const float* __restrict__ g, float* l) {
  const uint32_t lds_off = (uint32_t)(uintptr_t)(void*)l;
  const uint64_t gaddr   = (uint64_t)(uintptr_t)(const void*)g;
  asm volatile("global_load_async_to_lds_b128 %0, %1, off"
               :
               : "v"(lds_off), "v"(gaddr)
               : "memory");
}

__device__ __forceinline__ void wait_async_copies() {
#if defined(__has_builtin) && __has_builtin(__builtin_amdgcn_s_wait_asynccnt)
  __builtin_amdgcn_s_wait_asynccnt(0);
#else
  asm volatile("s_wait_asynccnt 0" ::: "memory");
#endif
}

__global__ __launch_bounds__(256)
void utou_persistent_wmma(const float* __restrict__ u,
                          const float* __restrict__ mask,
                          const float* __restrict__ normalizer,
                          const int*   __restrict__ coloring,
                          const float* __restrict__ W1,
                          const float* __restrict__ b1,
                          const float* __restrict__ W2,
                          const float* __restrict__ W3,
                          float* __restrict__ out,
                          int nNodes, int numGroups)
{
  __shared__ float w1_lds[kD * kStride];            // 67584 B, staged once per block
  __shared__ float u_lds[2][kRows * kStride];       // 2 x 67584 B, double-buffered
  __shared__ float mean2[kNPG * kD];
  __shared__ float z2s[kNPG * kD];
  __shared__ float z3s[kNPG * kD];
  __shared__ float b1_l[kD];
  __shared__ float mask_l[kRows];
  __shared__ int   col_l[kNPG];
  __shared__ float inv_n[kNPG];

  const int tid = threadIdx.x;
  const size_t totalRows = (size_t)nNodes * kC;

  // ---- stage W1 + b1 ONCE per persistent block ----
  for (int i = tid; i < kD * (kD / 4); i += 256) {
    const int row = i >> 5;
    const int c4  = (i & 31) << 2;
    *(float4*)&w1_lds[row * kStride + c4] = *(const float4*)(W1 + (size_t)row * kD + c4);
  }
  if (tid < kD) b1_l[tid] = b1[tid];

  if ((int)blockIdx.x >= numGroups) return;   // uniform per block

  const int lane = tid & 31;
  const int wave = tid >> 5;
  const int hi   = lane >> 4;      // half-wave: K offset +2 (A/B), row offset +8 (C/D)
  const int lm   = lane & 15;
  const int rBase = wave * 16;     // this wave's 16-row tile

  // ---- prologue: stage first group's u into buffer 0 (async) ----
  {
    const size_t rowBaseG = (size_t)blockIdx.x * kRows;
    for (int i = tid; i < kRows * (kD / 4); i += 256) {
      const int row = i >> 5;
      const int c4  = (i & 31) << 2;
      size_t gRow = rowBaseG + row;
      if (gRow >= totalRows) gRow = totalRows - 1;
      copy16_to_lds(u + gRow * kD + c4, &u_lds[0][row * kStride + c4]);
    }
  }
  wait_async_copies();
  __syncthreads();

  int pbuf = 0;
  for (int g = blockIdx.x; g < numGroups; g += gridDim.x, pbuf ^= 1) {
    const size_t rowBaseG = (size_t)g * kRows;
    const int n0 = g * kNPG;

    // ---- async-prefetch NEXT group's u into the other buffer (overlaps compute) ----
    const int gn = g + gridDim.x;
    if (gn < numGroups) {
      const size_t rowBaseN = (size_t)gn * kRows;
      for (int i = tid; i < kRows * (kD / 4); i += 256) {
        const int row = i >> 5;
        const int c4  = (i & 31) << 2;
        size_t gRow = rowBaseN + row;
        if (gRow >= totalRows) gRow = totalRows - 1;
        copy16_to_lds(u + gRow * kD + c4, &u_lds[pbuf ^ 1][row * kStride + c4]);
      }
    }

    // ---- per-group scalars ----
    if (tid < kRows) {
      const size_t gRow = rowBaseG + tid;
      mask_l[tid] = (gRow < totalRows) ? mask[gRow] : 0.0f;
    }
    if (tid < kNPG) {
      int n = n0 + tid; if (n >= nNodes) n = nNodes - 1;
      col_l[tid] = coloring[n];
      inv_n[tid] = 1.0f / normalizer[n];
    }
    __syncthreads();

    const float* __restrict__ ub = u_lds[pbuf];

    // ---- mean over colors ----
    for (int idx = tid; idx < kNPG * kD; idx += 256) {
      const int nl = idx >> 7;
      const int d  = idx & (kD - 1);
      float s = 0.0f;
      #pragma unroll
      for (int c = 0; c < kC; ++c) s += ub[(nl * kC + c) * kStride + d];
      mean2[idx] = s * inv_n[nl];
    }
    __syncthreads();

    // ---- z2 = 0.1*mean2@W2^T, z3 = 0.1*mean2@W3^T (W2/W3 are L2-resident) ----
    for (int idx = tid; idx < 2 * kNPG * kD; idx += 256) {
      const int mat = idx >> 9;
      const int rem = idx & 511;
      const int nl  = rem >> 7;
      const int e   = rem & (kD - 1);
      const float* __restrict__ Wr = (mat ? W3 : W2) + (size_t)e * kD;
      const float* __restrict__ m  = &mean2[nl * kD];
      float acc = 0.0f;
      #pragma unroll 4
      for (int d = 0; d < kD; d += 4) {
        const float4 w = *(const float4*)(Wr + d);
        acc += m[d] * w.x + m[d + 1] * w.y + m[d + 2] * w.z + m[d + 3] * w.w;
      }
      (mat ? z3s : z2s)[rem] = acc * 0.1f;
    }
    __syncthreads();

    // ---- main GEMM: two independent WMMA accumulation chains per tile ----
    const float* __restrict__ aPtr = &ub[(rBase + lm) * kStride + 2 * hi];
    for (int j = 0; j < 8; ++j) {
      const int e = j * 16 + lm;
      const float* __restrict__ bPtr = &w1_lds[e * kStride + 2 * hi];
      v8f acc0 = {};
      v8f acc1 = {};
      #pragma unroll
      for (int t = 0; t < 32; t += 2) {
        const v2f a0  = *(const v2f*)(aPtr + 4 * t);
        const v2f b0  = *(const v2f*)(bPtr + 4 * t);
        const v2f a1  = *(const v2f*)(aPtr + 4 * t + 4);
        const v2f b1f = *(const v2f*)(bPtr + 4 * t + 4);
        acc0 = __builtin_amdgcn_wmma_f32_16x16x4_f32(false, a0, false, b0,
                                                     (short)0, acc0, false, false);
        acc1 = __builtin_amdgcn_wmma_f32_16x16x4_f32(false, a1, false, b1f,
                                                     (short)0, acc1, false, false);
      }
      // ---- fused epilogue ----
      const float bias = b1_l[e];
      #pragma unroll
      for (int v = 0; v < 8; ++v) {
        const int r  = rBase + v + 8 * hi;
        const int nl = r >> 5;
        const int c  = r & (kC - 1);
        float val = acc0[v] + acc1[v]
                  + ub[r * kStride + e]          // residual from LDS
                  + bias
                  + z3s[nl * kD + e] * mask_l[r];
        if (c == col_l[nl]) val += z2s[nl * kD + e];
        const size_t gRow = rowBaseG + r;
        if (gRow < totalRows) out[gRow * kD + e] = val;
      }
    }

    wait_async_copies();   // own async prefetches done
    __syncthreads();       // everyone done reading buf + writing next buf
  }
}

extern "C" void kernel_launch(void* const* d_in, const int* in_sizes, int n_in,
                              void* d_out, int out_size, void* d_ws, size_t ws_size,
                              hipStream_t stream) {
  const float* u    = (const float*)d_in[0];
  const float* mask = (const float*)d_in[1];
  const float* norm = (const float*)d_in[2];
  const int*   col  = (const int*)d_in[3];
  const float* W1   = (const float*)d_in[4];
  const float* b1   = (const float*)d_in[5];
  const float* W2   = (const float*)d_in[6];
  const float* W3   = (const float*)d_in[7];
  float* out = (float*)d_out;

  const int nNodes = in_sizes[0] / (kC * kD);                 // 20000
  const int numGroups = (nNodes + kNPG - 1) / kNPG;           // 5000
  int blocks = numGroups < 625 ? numGroups : 625;             // persistent blocks
  utou_persistent_wmma<<<blocks, 256, 0, stream>>>(u, mask, norm, col, W1, b1, W2, W3,
                                                   out, nNodes, numGroups);
}